// PromptDetectionLoss_386547057293
// MI455X (gfx1250) — compile-verified
//
#include <hip/hip_runtime.h>
#include <hip/hip_bf16.h>
#include <stdint.h>

// Problem constants (fixed by the reference).
#define P_TOT   86016   // anchors: 256^2 + 128^2 + 64^2
#define G_TOT   100
#define G_PAD   112     // 7 WMMA N-tiles of 16
#define NT      7       // G_PAD / 16
#define C_TOT   365
#define TOPK_K  13
#define CAP     4096    // max candidates per GT (true max ~2.2K)
#define EPS_F   1e-7f
#define WAVES   8       // waves per block in the WMMA kernel

typedef __attribute__((ext_vector_type(2))) float v2f;
typedef __attribute__((ext_vector_type(8))) float v8f;

__device__ __forceinline__ float sigmoidf_(float x) { return 1.0f / (1.0f + expf(-x)); }

// ---------------------------------------------------------------------------
// K0: init workspace + preprocess GT boxes (centers, half-sizes, validity).
// ---------------------------------------------------------------------------
__global__ __launch_bounds__(256) void k_init(
    const float* __restrict__ gt_boxes, const int* __restrict__ gt_labels,
    const unsigned char* __restrict__ vmask,
    unsigned long long* __restrict__ best, int* __restrict__ hasC,
    int* __restrict__ candCount, float* __restrict__ gprep,
    int* __restrict__ labC, int* __restrict__ gvalid)
{
    int i = blockIdx.x * blockDim.x + threadIdx.x;
    if (i < P_TOT) best[i] = 0ull;
    if (i < G_PAD) {
        hasC[i] = 0;
        candCount[i] = 0;
        float x0 = 0.f, y0 = 0.f, x1 = 0.f, y1 = 0.f;
        int lab = -1, val = 0;
        if (i < G_TOT) {
            x0 = gt_boxes[i * 4 + 0]; y0 = gt_boxes[i * 4 + 1];
            x1 = gt_boxes[i * 4 + 2]; y1 = gt_boxes[i * 4 + 3];
            int l  = gt_labels[i];
            int lc = min(max(l, 0), C_TOT - 1);
            val = (l >= 0 && l < C_TOT && vmask[lc] != 0) ? 1 : 0;
            lab = lc;
        }
        gprep[0 * G_PAD + i] = x0;
        gprep[1 * G_PAD + i] = y0;
        gprep[2 * G_PAD + i] = x1;
        gprep[3 * G_PAD + i] = y1;
        gprep[4 * G_PAD + i] = (x0 + x1) * 0.5f;                 // cx
        gprep[5 * G_PAD + i] = (y0 + y1) * 0.5f;                 // cy
        gprep[6 * G_PAD + i] = fmaxf((x1 - x0) * 0.5f, 1.0f);    // hx
        gprep[7 * G_PAD + i] = fmaxf((y1 - y0) * 0.5f, 1.0f);    // hy
        labC[i]   = lab;
        gvalid[i] = val;
    }
}

// ---------------------------------------------------------------------------
// K1: has_center[g] = any anchor strictly (inside && center radius) in box g.
// ---------------------------------------------------------------------------
__global__ __launch_bounds__(256) void k_hascenter(
    const float* __restrict__ anch, const float* __restrict__ gprep,
    int* __restrict__ hasC)
{
    __shared__ float sg[8 * G_TOT];
    for (int j = threadIdx.x; j < 8 * G_TOT; j += blockDim.x) {
        int f = j / G_TOT, g = j % G_TOT;
        sg[j] = gprep[f * G_PAD + g];
    }
    __syncthreads();
    int r = blockIdx.x * blockDim.x + threadIdx.x;
    if (r >= P_TOT) return;
    float ax = anch[r * 2 + 0], ay = anch[r * 2 + 1];
    for (int g = 0; g < G_TOT; ++g) {
        float x0 = sg[0 * G_TOT + g], y0 = sg[1 * G_TOT + g];
        float x1 = sg[2 * G_TOT + g], y1 = sg[3 * G_TOT + g];
        if (ax >= x0 && ax <= x1 && ay >= y0 && ay <= y1) {
            float dx = fabsf(ax - sg[4 * G_TOT + g]) / sg[6 * G_TOT + g];
            float dy = fabsf(ay - sg[5 * G_TOT + g]) / sg[7 * G_TOT + g];
            if (fmaxf(dx, dy) <= 0.5f) hasC[g] = 1;  // idempotent store, race-free
        }
    }
}

// ---------------------------------------------------------------------------
// K2: WMMA one-hot gather of pred_scores[p, label[g]] fused with the
//     alignment metric; appends candidates (cand && valid) to per-GT lists.
//     One wave = 16-anchor tile x 112 GTs, f32 16x16x4 WMMA over K=C.
//     Wave-uniform __any skip: WMMA issued only for K-chunks where some lane's
//     label hits (exact: skipped WMMAs have an all-zero B). EXEC stays all-1s
//     (scalar branch), satisfying the WMMA EXEC restriction.
// ---------------------------------------------------------------------------
__global__ __launch_bounds__(256) void k_align(
    const float* __restrict__ PS, const float* __restrict__ pboxes,
    const float* __restrict__ pobj, const float* __restrict__ anch,
    const float* __restrict__ gprep, const int* __restrict__ labC,
    const int* __restrict__ gvalid, const int* __restrict__ hasC,
    int* __restrict__ candCount, float* __restrict__ candVal,
    int* __restrict__ candIdx)
{
    __shared__ float sAnc[WAVES][16][8];  // ax,ay,px0,py0,px1,py1,obj_sig,pad

    const int wave = threadIdx.x >> 5;
    const int lane = threadIdx.x & 31;
    const int tile = blockIdx.x * WAVES + wave;
    const int ab   = tile * 16;           // anchor base of this wave's tile
    const int hi   = lane >> 4;
    const int n    = lane & 15;

    // Stage the 16 anchors' data (shared by both half-waves) into LDS.
    if (lane < 16) {
        int r = ab + lane;
        sAnc[wave][lane][0] = anch[r * 2 + 0];
        sAnc[wave][lane][1] = anch[r * 2 + 1];
        sAnc[wave][lane][2] = pboxes[r * 4 + 0];
        sAnc[wave][lane][3] = pboxes[r * 4 + 1];
        sAnc[wave][lane][4] = pboxes[r * 4 + 2];
        sAnc[wave][lane][5] = pboxes[r * 4 + 3];
        sAnc[wave][lane][6] = sigmoidf_(pobj[r]);
        sAnc[wave][lane][7] = 0.0f;
    }
    __syncthreads();

    // Each lane needs the label of GT column n in every N-tile (one-hot B).
    int labs[NT];
#pragma unroll
    for (int t = 0; t < NT; ++t) labs[t] = labC[t * 16 + n];

    v8f zero = {0.f, 0.f, 0.f, 0.f, 0.f, 0.f, 0.f, 0.f};
    v8f acc[NT];
#pragma unroll
    for (int t = 0; t < NT; ++t) acc[t] = zero;

    // A fragment row for this lane (M = lane & 15); streams its score row.
    const float* rowp = PS + (size_t)(ab + n) * C_TOT;

    // Main K loop: K = 0..363 in steps of 4 (91 iters, no bounds checks).
    for (int k0 = 0; k0 <= 360; k0 += 4) {
        int kA = k0 + 2 * hi;
        v2f a; a.x = rowp[kA]; a.y = rowp[kA + 1];
#pragma unroll
        for (int t = 0; t < NT; ++t) {
            bool hit = ((unsigned)(labs[t] - k0)) < 4u;  // label in [k0, k0+4)
            if (__any(hit)) {
                v2f b;
                b.x = (labs[t] == kA)     ? 1.0f : 0.0f;
                b.y = (labs[t] == kA + 1) ? 1.0f : 0.0f;
                acc[t] = __builtin_amdgcn_wmma_f32_16x16x4_f32(
                    false, a, false, b, (short)0, acc[t], false, false);
            }
        }
    }
    // Tail: K = 364 (cols 365..367 are zero padding).
    {
        v2f a; a.x = (hi == 0) ? rowp[364] : 0.0f; a.y = 0.0f;
#pragma unroll
        for (int t = 0; t < NT; ++t) {
            bool hit = (labs[t] == 364);
            if (__any(hit)) {
                v2f b;
                b.x = (hi == 0 && labs[t] == 364) ? 1.0f : 0.0f;
                b.y = 0.0f;
                acc[t] = __builtin_amdgcn_wmma_f32_16x16x4_f32(
                    false, a, false, b, (short)0, acc[t], false, false);
            }
        }
    }

    // Epilogue: per (g, p) alignment + candidate append.
#pragma unroll
    for (int t = 0; t < NT; ++t) {
        int g = t * 16 + n;
        float x0 = gprep[0 * G_PAD + g], y0 = gprep[1 * G_PAD + g];
        float x1 = gprep[2 * G_PAD + g], y1 = gprep[3 * G_PAD + g];
        float cx = gprep[4 * G_PAD + g], cy = gprep[5 * G_PAD + g];
        float hx = gprep[6 * G_PAD + g], hy = gprep[7 * G_PAD + g];
        int   vg = gvalid[g];
        int   hc = hasC[g];
        float areag = fmaxf(x1 - x0, 0.f) * fmaxf(y1 - y0, 0.f);
#pragma unroll
        for (int v = 0; v < 8; ++v) {
            int   m   = v + 8 * hi;            // C/D layout: M = v + 8*hi
            float ax  = sAnc[wave][m][0], ay  = sAnc[wave][m][1];
            float px0 = sAnc[wave][m][2], py0 = sAnc[wave][m][3];
            float px1 = sAnc[wave][m][4], py1 = sAnc[wave][m][5];
            float objs = sAnc[wave][m][6];
            bool inside = ax >= x0 && ax <= x1 && ay >= y0 && ay <= y1;
            float dx = fabsf(ax - cx) / hx;
            float dy = fabsf(ay - cy) / hy;
            bool strict = inside && (fmaxf(dx, dy) <= 0.5f);
            bool cand   = hc ? strict : inside;
            if (cand && vg) {
                float ltx = fmaxf(x0, px0), lty = fmaxf(y0, py0);
                float rbx = fminf(x1, px1), rby = fminf(y1, py1);
                float inter = fmaxf(rbx - ltx, 0.f) * fmaxf(rby - lty, 0.f);
                float areap = fmaxf(px1 - px0, 0.f) * fmaxf(py1 - py0, 0.f);
                float iou = inter / (areag + areap - inter + EPS_F);
                float cls = sigmoidf_(acc[t][v]);          // exact f32 gather
                float q   = sqrtf(fmaxf(objs * cls, 0.f));
                float prior = expf(-0.5f * (dx * dx + dy * dy));
                float i2 = iou * iou;
                float i6 = i2 * i2 * i2;
                float align = q * i6 * prior;
                int slot = atomicAdd(&candCount[g], 1);
                if (slot < CAP) {
                    candVal[g * CAP + slot] = align;
                    candIdx[g * CAP + slot] = ab + m;
                }
            }
        }
    }
}

// ---------------------------------------------------------------------------
// K3: per-GT exact top-13 over its candidate list; scatter winners into the
//     per-anchor packed (value, ~g) atomicMax — reproduces sel.max/argmax
//     with jnp's lowest-index tie-break.
// ---------------------------------------------------------------------------
__global__ __launch_bounds__(256) void k_topk(
    const float* __restrict__ candVal, const int* __restrict__ candIdx,
    const int* __restrict__ candCount, unsigned long long* __restrict__ best)
{
    const int g   = blockIdx.x;
    const int tid = threadIdx.x;
    int nc = candCount[g];
    if (nc > CAP) nc = CAP;

    float lv[CAP / 256];
    int   li[CAP / 256];
#pragma unroll
    for (int j = 0; j < CAP / 256; ++j) {
        int i = tid + j * 256;
        if (i < nc) { lv[j] = candVal[g * CAP + i]; li[j] = candIdx[g * CAP + i]; }
        else        { lv[j] = -1.0f;                li[j] = 0; }
    }

    __shared__ unsigned long long red[256];
    for (int r = 0; r < TOPK_K; ++r) {
        unsigned long long mypack = 0ull;
#pragma unroll
        for (int j = 0; j < CAP / 256; ++j) {
            if (lv[j] >= 0.0f) {
                unsigned long long p =
                    ((unsigned long long)__float_as_uint(lv[j]) << 32) |
                    (unsigned long long)(0xFFFFFFFFu - (unsigned)li[j]);
                if (p > mypack) mypack = p;
            }
        }
        red[tid] = mypack;
        __syncthreads();
        for (int s = 128; s > 0; s >>= 1) {
            if (tid < s) {
                unsigned long long o = red[tid + s];
                if (o > red[tid]) red[tid] = o;
            }
            __syncthreads();
        }
        unsigned long long win = red[0];
        __syncthreads();
        if (win != 0ull) {
            unsigned widx = 0xFFFFFFFFu - (unsigned)(win & 0xFFFFFFFFull);
#pragma unroll
            for (int j = 0; j < CAP / 256; ++j)
                if (lv[j] >= 0.0f && (unsigned)li[j] == widx) lv[j] = -1.0f;
            if (tid == 0) {
                unsigned long long pb = (win & 0xFFFFFFFF00000000ull) |
                    (unsigned long long)(0xFFFFFFFFu - (unsigned)g);
                atomicMax(&best[widx], pb);
            }
        }
    }
}

// ---------------------------------------------------------------------------
// K4: final outputs. One wave per anchor: writes its full 365-wide score row
//     (fused zero-fill + sparse set), boxes, fg, matched indices/labels.
// ---------------------------------------------------------------------------
__global__ __launch_bounds__(256) void k_final(
    const float* __restrict__ pboxes, const float* __restrict__ gprep,
    const int* __restrict__ labC, const unsigned long long* __restrict__ best,
    float* __restrict__ out_scores, float* __restrict__ out_boxes,
    float* __restrict__ out_fg, float* __restrict__ out_mgi,
    float* __restrict__ out_mlab)
{
    const int wave = threadIdx.x >> 5;
    const int lane = threadIdx.x & 31;
    const int r = blockIdx.x * 8 + wave;

    unsigned long long b = best[r];
    bool fg = (b != 0ull);
    int g = fg ? (int)(0xFFFFFFFFu - (unsigned)(b & 0xFFFFFFFFull)) : 0;

    float x0 = gprep[0 * G_PAD + g], y0 = gprep[1 * G_PAD + g];
    float x1 = gprep[2 * G_PAD + g], y1 = gprep[3 * G_PAD + g];

    float sv = 0.0f;
    int lab = 0;
    if (fg) {
        lab = labC[g];
        float px0 = pboxes[r * 4 + 0], py0 = pboxes[r * 4 + 1];
        float px1 = pboxes[r * 4 + 2], py1 = pboxes[r * 4 + 3];
        float ltx = fmaxf(x0, px0), lty = fmaxf(y0, py0);
        float rbx = fminf(x1, px1), rby = fminf(y1, py1);
        float inter = fmaxf(rbx - ltx, 0.f) * fmaxf(rby - lty, 0.f);
        float areag = fmaxf(x1 - x0, 0.f) * fmaxf(y1 - y0, 0.f);
        float areap = fmaxf(px1 - px0, 0.f) * fmaxf(py1 - py0, 0.f);
        float iou = inter / (areag + areap - inter + EPS_F);
        sv = fmaxf(iou, 0.1f);  // clip(overlap, 0.1)
    }

    size_t base = (size_t)r * C_TOT;
    for (int j = lane; j < C_TOT; j += 32)
        out_scores[base + j] = (fg && j == lab) ? sv : 0.0f;

    if (lane == 0) {
        out_boxes[r * 4 + 0] = fg ? x0 : 0.0f;
        out_boxes[r * 4 + 1] = fg ? y0 : 0.0f;
        out_boxes[r * 4 + 2] = fg ? x1 : 0.0f;
        out_boxes[r * 4 + 3] = fg ? y1 : 0.0f;
        out_fg[r]   = fg ? 1.0f : 0.0f;
        out_mgi[r]  = fg ? (float)g : -1.0f;
        out_mlab[r] = fg ? (float)lab : -1.0f;
    }
}

// ---------------------------------------------------------------------------
extern "C" void kernel_launch(void* const* d_in, const int* in_sizes, int n_in,
                              void* d_out, int out_size, void* d_ws, size_t ws_size,
                              hipStream_t stream) {
    (void)in_sizes; (void)n_in; (void)out_size; (void)ws_size;
    const float*         PS     = (const float*)d_in[0];          // (P,C)
    const float*         pboxes = (const float*)d_in[1];          // (P,4)
    const float*         pobj   = (const float*)d_in[2];          // (P,)
    const float*         anch   = (const float*)d_in[3];          // (P,2)
    const float*         gtb    = (const float*)d_in[4];          // (G,4)
    const int*           glab   = (const int*)d_in[5];            // (G,)
    const unsigned char* vmask  = (const unsigned char*)d_in[6];  // (C,) bool

    // Workspace layout (256B aligned slabs), ~4.4 MB total.
    char* ws = (char*)d_ws;
    size_t off = 0;
    auto take = [&](size_t bytes) { char* p = ws + off; off = (off + bytes + 255) & ~(size_t)255; return p; };
    unsigned long long* best  = (unsigned long long*)take((size_t)P_TOT * 8);
    float* candVal            = (float*)take((size_t)G_PAD * CAP * 4);
    int*   candIdx            = (int*)  take((size_t)G_PAD * CAP * 4);
    float* gprep              = (float*)take((size_t)8 * G_PAD * 4);
    int*   labC               = (int*)  take((size_t)G_PAD * 4);
    int*   gvalid             = (int*)  take((size_t)G_PAD * 4);
    int*   hasC               = (int*)  take((size_t)G_PAD * 4);
    int*   candCount          = (int*)  take((size_t)G_PAD * 4);

    float* out = (float*)d_out;
    float* out_scores = out;
    float* out_boxes  = out_scores + (size_t)P_TOT * C_TOT;
    float* out_fg     = out_boxes  + (size_t)P_TOT * 4;
    float* out_mgi    = out_fg     + P_TOT;
    float* out_mlab   = out_mgi    + P_TOT;

    k_init<<<(P_TOT + 255) / 256, 256, 0, stream>>>(
        gtb, glab, vmask, best, hasC, candCount, gprep, labC, gvalid);
    k_hascenter<<<(P_TOT + 255) / 256, 256, 0, stream>>>(anch, gprep, hasC);
    k_align<<<P_TOT / (16 * WAVES), 256, 0, stream>>>(
        PS, pboxes, pobj, anch, gprep, labC, gvalid, hasC,
        candCount, candVal, candIdx);
    k_topk<<<G_TOT, 256, 0, stream>>>(candVal, candIdx, candCount, best);
    k_final<<<P_TOT / 8, 256, 0, stream>>>(
        pboxes, gprep, labC, best, out_scores, out_boxes, out_fg, out_mgi, out_mlab);
}